// VariableEmbedding_9199819948534
// MI455X (gfx1250) — compile-verified
//
#include <hip/hip_runtime.h>
#include <cstdint>

#define EPSV 1e-5f

typedef float        f32x4 __attribute__((ext_vector_type(4)));
typedef unsigned int u32x4 __attribute__((ext_vector_type(4)));
typedef int          i32x4 __attribute__((ext_vector_type(4)));
typedef int          i32x8 __attribute__((ext_vector_type(8)));

#if !__has_builtin(__builtin_amdgcn_tensor_load_to_lds)
#define NO_TDM 1
#endif

// ---------------------------------------------------------------------------
// Pass 1: per-variable stats. One wave (32 lanes) per v.
//   ws[0      .. 4096)  : W' = W - mean_d(W)      [V=32][D=128]
//   ws[4096   .. 8192)  : b' = b - mean_d(b)      [V=32][D=128]
//   ws[8192   .. 8320)  : per-v {mean(W'^2), mean(W'b'), mean(b'^2), pad}
// ---------------------------------------------------------------------------
__global__ __launch_bounds__(32) void vle_stats_kernel(
    const float* __restrict__ W, const float* __restrict__ b,
    float* __restrict__ ws) {
  const int v = blockIdx.x;        // 0..31
  const int lane = threadIdx.x;    // 0..31
  const f32x4* W4 = (const f32x4*)(W + v * 128);
  const f32x4* B4 = (const f32x4*)(b + v * 128);
  f32x4 w = W4[lane];
  f32x4 bb = B4[lane];

  float sw = w.x + w.y + w.z + w.w;
  float sb = bb.x + bb.y + bb.z + bb.w;
  for (int o = 16; o > 0; o >>= 1) {
    sw += __shfl_xor(sw, o, 32);
    sb += __shfl_xor(sb, o, 32);
  }
  const float Wm = sw * (1.0f / 128.0f);
  const float bm = sb * (1.0f / 128.0f);

  f32x4 wc; wc.x = w.x - Wm; wc.y = w.y - Wm; wc.z = w.z - Wm; wc.w = w.w - Wm;
  f32x4 bc; bc.x = bb.x - bm; bc.y = bb.y - bm; bc.z = bb.z - bm; bc.w = bb.w - bm;

  float a = wc.x*wc.x + wc.y*wc.y + wc.z*wc.z + wc.w*wc.w;
  float c = wc.x*bc.x + wc.y*bc.y + wc.z*bc.z + wc.w*bc.w;
  float e = bc.x*bc.x + bc.y*bc.y + bc.z*bc.z + bc.w*bc.w;
  for (int o = 16; o > 0; o >>= 1) {
    a += __shfl_xor(a, o, 32);
    c += __shfl_xor(c, o, 32);
    e += __shfl_xor(e, o, 32);
  }

  ((f32x4*)ws)[v * 32 + lane]          = wc;
  ((f32x4*)(ws + 4096))[v * 32 + lane] = bc;
  if (lane == 0) {
    ws[8192 + 4 * v + 0] = a * (1.0f / 128.0f);
    ws[8192 + 4 * v + 1] = c * (1.0f / 128.0f);
    ws[8192 + 4 * v + 2] = e * (1.0f / 128.0f);
    ws[8192 + 4 * v + 3] = 0.0f;
  }
}

// ---------------------------------------------------------------------------
// TDM: 1-D tensor load into LDS (data_size = 4 bytes, n <= 65535 elements).
// Descriptor bitfields per CDNA5 ISA ch.8 (D# group0 / group1).
// clang-23 toolchain: 6-arg builtin (g0, g1, g2, g3, extra group, cpol).
// ---------------------------------------------------------------------------
#ifndef NO_TDM
__device__ __forceinline__ void tdm_load_1d(uint32_t lds_off,
                                            const void* gptr, uint32_t n) {
  const uint64_t ga = (uint64_t)(uintptr_t)gptr;
  u32x4 g0;
  g0[0] = 1u;                                    // count=1 (valid), user mode
  g0[1] = lds_off;                               // lds_addr [63:32]
  g0[2] = (uint32_t)ga;                          // global_addr low
  g0[3] = ((uint32_t)(ga >> 32) & 0x01FFFFFFu)   // global_addr[56:32]
          | (2u << 30);                          // type = 2 ("image")
  i32x8 g1;
  g1[0] = (int)(2u << 16);                       // data_size = 2 -> 4 bytes
  g1[1] = (int)((n & 0xFFFFu) << 16);            // tensor_dim0 low16 @ [63:48]
  g1[2] = (int)((n >> 16) & 0xFFFFu);            // tensor_dim0 hi16; dim1=0
  g1[3] = (int)((n & 0xFFFFu) << 16);            // tile_dim0 @ [127:112]
  g1[4] = 0;                                     // tile_dim1=0, tile_dim2=0
  g1[5] = (int)n;                                // tensor_dim0_stride low
  g1[6] = 0;
  g1[7] = 0;
  i32x4 z4 = {0, 0, 0, 0};
  i32x8 z8 = {0, 0, 0, 0, 0, 0, 0, 0};
  __builtin_amdgcn_tensor_load_to_lds(g0, g1, z4, z4, z8, 0);
}
#endif

// ---------------------------------------------------------------------------
// Pass 2: streaming fused kernel. One wave = one 128-elem output row per
// iteration; 64 contiguous rows per wave; float4 non-temporal stores.
// LDS layout (floats): W'[0,4096) b'[4096,8192) stats[8192,8320)
//                      gamma[8320,12416) beta[12416,16512)
// ---------------------------------------------------------------------------
__global__ __launch_bounds__(256) void vle_fused_kernel(
    const float* __restrict__ x, const float* __restrict__ gamma,
    const float* __restrict__ beta, const float* __restrict__ wsblk,
    float* __restrict__ out, long long rows) {
  __shared__ __attribute__((aligned(16))) float sh[16512];

#ifndef NO_TDM
  if (threadIdx.x < 32) {  // wave 0 only; TDM issue ignores EXEC
    const uint32_t shbase = (uint32_t)(uintptr_t)(&sh[0]);
    tdm_load_1d(shbase,             wsblk, 8320u);
    tdm_load_1d(shbase + 8320u*4u,  gamma, 4096u);
    tdm_load_1d(shbase + 12416u*4u, beta,  4096u);
    __builtin_amdgcn_s_wait_tensorcnt(0);
  }
#else
  for (int idx = threadIdx.x; idx < 8320; idx += 256) sh[idx] = wsblk[idx];
  for (int idx = threadIdx.x; idx < 4096; idx += 256) {
    sh[8320 + idx]  = gamma[idx];
    sh[12416 + idx] = beta[idx];
  }
#endif
  __syncthreads();

  const int lane = threadIdx.x & 31;
  const int wave = threadIdx.x >> 5;
  const long long waveId = (long long)blockIdx.x * 8 + wave;
  const long long base = waveId * 64;

  const f32x4* __restrict__ shWc4 = (const f32x4*)&sh[0];
  const f32x4* __restrict__ shBc4 = (const f32x4*)&sh[4096];
  const float* __restrict__ shSt  = &sh[8192];
  const f32x4* __restrict__ shG4  = (const f32x4*)&sh[8320];
  const f32x4* __restrict__ shBe4 = (const f32x4*)&sh[12416];

  for (int g = 0; g < 2; ++g) {
    const long long rbase = base + (long long)g * 32;  // multiple of 32 -> v==i
    if (rbase >= rows) break;
    float xv = 0.0f;
    if (rbase + lane < rows) xv = x[rbase + lane];  // coalesced, 1 per 32 rows

#pragma unroll 8
    for (int i = 0; i < 32; ++i) {
      const long long r = rbase + i;
      if (r >= rows) break;
      const float xs = __shfl(xv, i, 32);  // broadcast row scalar
      const int v = i;

      const float sA = shSt[4 * v + 0];
      const float sC = shSt[4 * v + 1];
      const float sE = shSt[4 * v + 2];
      // var = sA*x^2 + 2*sC*x + sE   (analytic LayerNorm variance)
      const float var = fmaf(fmaf(sA, xs, 2.0f * sC), xs, sE);
      const float rinv = rsqrtf(var + EPSV);

      const int p = v * 32 + lane;
      const f32x4 wc = shWc4[p];
      const f32x4 bc = shBc4[p];
      const f32x4 gg = shG4[p];
      const f32x4 be = shBe4[p];

      f32x4 rg; rg.x = rinv * gg.x; rg.y = rinv * gg.y;
                rg.z = rinv * gg.z; rg.w = rinv * gg.w;
      f32x4 o;
      o.x = fmaxf(0.0f, fmaf(fmaf(xs, wc.x, bc.x), rg.x, be.x));
      o.y = fmaxf(0.0f, fmaf(fmaf(xs, wc.y, bc.y), rg.y, be.y));
      o.z = fmaxf(0.0f, fmaf(fmaf(xs, wc.z, bc.z), rg.z, be.z));
      o.w = fmaxf(0.0f, fmaf(fmaf(xs, wc.w, bc.w), rg.w, be.w));

      f32x4* dst = (f32x4*)(out + (size_t)r * 128 + (size_t)(lane << 2));
      __builtin_nontemporal_store(o, dst);  // streaming 512MB: bypass cache
    }
  }
}

// ---------------------------------------------------------------------------
extern "C" void kernel_launch(void* const* d_in, const int* in_sizes, int n_in,
                              void* d_out, int out_size, void* d_ws,
                              size_t ws_size, hipStream_t stream) {
  (void)n_in; (void)out_size; (void)ws_size;
  const float* x     = (const float*)d_in[0];  // [B,S,V]
  const float* W     = (const float*)d_in[1];  // [V,D]
  const float* b     = (const float*)d_in[2];  // [V,D]
  const float* gamma = (const float*)d_in[3];  // [V,D]
  const float* beta  = (const float*)d_in[4];  // [V,D]
  float* out = (float*)d_out;
  float* ws  = (float*)d_ws;  // needs 8320 floats (33,280 B)

  const long long rows = in_sizes[0];  // B*S*V; one row = D=128 outputs

  vle_stats_kernel<<<32, 32, 0, stream>>>(W, b, ws);

  // 8 waves/block * 64 rows/wave = 512 rows per block
  const int nblocks = (int)((rows + 511) / 512);
  vle_fused_kernel<<<nblocks, 256, 0, stream>>>(x, gamma, beta, ws, out, rows);
}